// gmodel01_44195213476163
// MI455X (gfx1250) — compile-verified
//
#include <hip/hip_runtime.h>
#include <hip/hip_bf16.h>

typedef float v2f __attribute__((ext_vector_type(2)));
typedef float v8f __attribute__((ext_vector_type(8)));

#define BB   256   // batch
#define MM   8     // metapaths
#define LL   64    // walk length
#define EE   128   // embedding
#define HH   8     // heads
#define HD   16    // head dim
#define E3   384   // 3*E

// ---------------------------------------------------------------------------
// Stage 1: gather node/edge embeddings, elementwise triple product, mean over L.
// One wave32 per (b,m) pair; each lane owns 4 embedding dims (32*4 = 128).
// ---------------------------------------------------------------------------
__global__ void k_gather(const int* __restrict__ walks,
                         const float* __restrict__ node_table,
                         const float* __restrict__ fc1_b,
                         const float* __restrict__ edge_table,
                         float* __restrict__ agg0)
{
    const int wavesPerBlock = blockDim.x >> 5;
    const int waveId = threadIdx.x >> 5;
    const int lane   = threadIdx.x & 31;
    const int pair   = blockIdx.x * wavesPerBlock + waveId;   // 0 .. B*M-1
    if (pair >= BB * MM) return;

    const int dim = lane * 4;
    float b0 = fc1_b[dim + 0];
    float b1 = fc1_b[dim + 1];
    float b2 = fc1_b[dim + 2];
    float b3 = fc1_b[dim + 3];

    float a0 = 0.f, a1 = 0.f, a2 = 0.f, a3 = 0.f;
    const int* wp = walks + (size_t)pair * LL * 3;

    for (int l = 0; l < LL; ++l) {
        // prefetch next step's indices into cache (global_prefetch_b8)
        __builtin_prefetch(wp + (l + 1) * 3, 0, 0);
        const int n1 = wp[l * 3 + 0];
        const int n2 = wp[l * 3 + 1];
        const int et = wp[l * 3 + 2];
        const float4 v1 = *(const float4*)(node_table + (size_t)n1 * EE + dim);
        const float4 v2 = *(const float4*)(node_table + (size_t)n2 * EE + dim);
        const float4 ve = *(const float4*)(edge_table + (size_t)et * EE + dim);
        a0 += (v1.x + b0) * (v2.x + b0) * ve.x;
        a1 += (v1.y + b1) * (v2.y + b1) * ve.y;
        a2 += (v1.z + b2) * (v2.z + b2) * ve.z;
        a3 += (v1.w + b3) * (v2.w + b3) * ve.w;
    }
    const float inv = 1.0f / (float)LL;
    float4 r;
    r.x = a0 * inv; r.y = a1 * inv; r.z = a2 * inv; r.w = a3 * inv;
    *(float4*)(agg0 + (size_t)pair * EE + dim) = r;          // [B,M,E]
}

// ---------------------------------------------------------------------------
// Generic fp32 WMMA 16x16 tile GEMM:  C[16,16] = A[16,K] * B[K,16] + bias
//   A: row-major with row stride lda
//   B: element (k,n) at Bm[k*bKs + n*bNs]  (handles both W and W^T)
// A-frag (32-bit 16x4): lanes 0-15 hold K=kk,kk+1; lanes 16-31 hold K=kk+2,kk+3.
// C/D frag: VGPR r -> row rowBase + r + 8*(lane>=16), col colBase + (lane&15).
// ---------------------------------------------------------------------------
__device__ __forceinline__ void gemm_tile_f32(
    const float* __restrict__ A, int lda,
    const float* __restrict__ Bm, int bKs, int bNs,
    const float* __restrict__ bias,
    float* __restrict__ C, int ldc,
    int rowBase, int colBase, int K)
{
    const int lane  = threadIdx.x & 31;
    const int laneM = lane & 15;
    const int hi    = lane >> 4;

    v8f acc = {};
    const float* arow = A + (size_t)(rowBase + laneM) * lda;
    for (int kk = 0; kk < K; kk += 4) {
        const int ka = kk + hi * 2;
        v2f a;
        a.x = arow[ka + 0];
        a.y = arow[ka + 1];
        v2f b;
        b.x = Bm[(size_t)(ka + 0) * bKs + (size_t)(colBase + laneM) * bNs];
        b.y = Bm[(size_t)(ka + 1) * bKs + (size_t)(colBase + laneM) * bNs];
        acc = __builtin_amdgcn_wmma_f32_16x16x4_f32(
            /*neg_a=*/false, a, /*neg_b=*/false, b,
            /*c_mod=*/(short)0, acc, /*reuse_a=*/false, /*reuse_b=*/false);
    }
    const float bv = bias ? bias[colBase + laneM] : 0.0f;
#pragma unroll
    for (int r = 0; r < 8; ++r) {
        const int row = rowBase + r + 8 * hi;
        C[(size_t)row * ldc + colBase + laneM] = acc[r] + bv;
    }
}

// ---------------------------------------------------------------------------
// Stage 2: per-metapath linear. out[m,b,:] = agg0[b,m,:] @ W_mp[sel(m)] + b_mp.
// Output written in [M,B,E] layout (seq-first), ready for the MHA GEMMs.
// Waves: m(8) x rowTiles(16) x colTiles(8) = 1024.
// ---------------------------------------------------------------------------
__global__ void k_mp_linear(const float* __restrict__ agg0,
                            const int* __restrict__ mp_sel,
                            const float* __restrict__ Wmp,
                            const float* __restrict__ bmp,
                            float* __restrict__ agg1)
{
    const int w = blockIdx.x * (blockDim.x >> 5) + (threadIdx.x >> 5);
    const int m = w / 128;           // 16*8 tiles per metapath
    const int t = w % 128;
    const int rowTile = t / 8;       // over B=256
    const int colTile = t % 8;       // over E=128
    const int sel = mp_sel[m];

    gemm_tile_f32(agg0 + (size_t)m * EE, MM * EE,          // A row b: agg0[(b*M+m)*E + k]
                  Wmp + (size_t)sel * EE * EE, EE, 1,      // B[k,n] = W[k*E+n]
                  bmp + (size_t)sel * EE,
                  agg1 + (size_t)m * BB * EE, EE,          // C row b -> agg1[(m*B+b)*E]
                  rowTile * 16, colTile * 16, EE);
}

// ---------------------------------------------------------------------------
// Stage 3a: QKV projection. qkv[r,:] = x[r,:] @ Wqkv^T + bqkv, r = m*B+b.
// Waves: rowTiles(128) x colTiles(24) = 3072.
// ---------------------------------------------------------------------------
__global__ void k_qkv(const float* __restrict__ agg1,
                      const float* __restrict__ Wqkv,
                      const float* __restrict__ bqkv,
                      float* __restrict__ qkv)
{
    const int w = blockIdx.x * (blockDim.x >> 5) + (threadIdx.x >> 5);
    const int rowTile = w / 24;
    const int colTile = w % 24;
    gemm_tile_f32(agg1, EE,
                  Wqkv, 1, EE,                  // B[k,n] = Wqkv[n*E + k]  (W^T)
                  bqkv,
                  qkv, E3,
                  rowTile * 16, colTile * 16, EE);
}

// ---------------------------------------------------------------------------
// Stage 3b: attention over metapaths. One thread per (b,h,m) output row.
// M=8, head dim=16: tiny — scalar VALU kernel.
// ---------------------------------------------------------------------------
__global__ void k_attn(const float* __restrict__ qkv,
                       float* __restrict__ attn_o)
{
    const int t = blockIdx.x * blockDim.x + threadIdx.x;
    if (t >= BB * HH * MM) return;
    const int m = t % MM;
    const int h = (t / MM) % HH;
    const int b = t / (MM * HH);

    const float scale = 0.25f;  // 1/sqrt(16)
    float q[HD];
    const float* qp = qkv + (size_t)(m * BB + b) * E3 + h * HD;
#pragma unroll
    for (int d = 0; d < HD; ++d) q[d] = qp[d];

    float s[MM];
    float mx = -3.0e38f;
#pragma unroll
    for (int n = 0; n < MM; ++n) {
        const float* kp = qkv + (size_t)(n * BB + b) * E3 + EE + h * HD;
        float acc = 0.f;
#pragma unroll
        for (int d = 0; d < HD; ++d) acc += q[d] * kp[d];
        acc *= scale;
        s[n] = acc;
        mx = fmaxf(mx, acc);
    }
    float sum = 0.f;
#pragma unroll
    for (int n = 0; n < MM; ++n) { s[n] = __expf(s[n] - mx); sum += s[n]; }
    const float rs = 1.0f / sum;

    float o[HD];
#pragma unroll
    for (int d = 0; d < HD; ++d) o[d] = 0.f;
#pragma unroll
    for (int n = 0; n < MM; ++n) {
        const float p = s[n] * rs;
        const float* vp = qkv + (size_t)(n * BB + b) * E3 + 2 * EE + h * HD;
#pragma unroll
        for (int d = 0; d < HD; ++d) o[d] += p * vp[d];
    }
    float* op = attn_o + (size_t)(m * BB + b) * EE + h * HD;
#pragma unroll
    for (int d = 0; d < HD; ++d) op[d] = o[d];
}

// ---------------------------------------------------------------------------
// Stage 3c: output projection. out[r,:] = o[r,:] @ Wo^T + bo.  [M,B,E] order.
// Waves: 128 x 8 = 1024.
// ---------------------------------------------------------------------------
__global__ void k_oproj(const float* __restrict__ attn_o,
                        const float* __restrict__ Wo,
                        const float* __restrict__ bo,
                        float* __restrict__ out)
{
    const int w = blockIdx.x * (blockDim.x >> 5) + (threadIdx.x >> 5);
    const int rowTile = w / 8;
    const int colTile = w % 8;
    gemm_tile_f32(attn_o, EE,
                  Wo, 1, EE,                    // B[k,n] = Wo[n*E + k]
                  bo,
                  out, EE,
                  rowTile * 16, colTile * 16, EE);
}

// ---------------------------------------------------------------------------
extern "C" void kernel_launch(void* const* d_in, const int* in_sizes, int n_in,
                              void* d_out, int out_size, void* d_ws, size_t ws_size,
                              hipStream_t stream)
{
    // inputs (setup_inputs order)
    // 0: node_idx (unused by reference)
    const int*   mp_type_idx = (const int*)  d_in[1];
    const int*   walks       = (const int*)  d_in[2];
    const float* node_table  = (const float*)d_in[3];
    const float* fc1_b       = (const float*)d_in[4];
    const float* edge_table  = (const float*)d_in[5];
    const float* W_mp        = (const float*)d_in[6];
    const float* b_mp        = (const float*)d_in[7];
    const float* Wqkv        = (const float*)d_in[8];
    const float* bqkv        = (const float*)d_in[9];
    const float* Wo          = (const float*)d_in[10];
    const float* bo          = (const float*)d_in[11];
    float* out = (float*)d_out;

    // workspace layout (floats)
    float* ws    = (float*)d_ws;
    float* agg0  = ws;                                   // [B,M,E]      262144
    float* agg1  = agg0 + (size_t)BB * MM * EE;          // [M,B,E]      262144
    float* qkvb  = agg1 + (size_t)BB * MM * EE;          // [M,B,3E]     786432
    float* attno = qkvb + (size_t)MM * BB * E3;          // [M,B,E]      262144

    // Stage 1: gather + fuse + mean.  2048 waves, 8 waves/block.
    k_gather<<<(BB * MM) / 8, 256, 0, stream>>>(walks, node_table, fc1_b,
                                                edge_table, agg0);

    // Stage 2: per-metapath linear. 1024 tile-waves.
    k_mp_linear<<<1024 / 8, 256, 0, stream>>>(agg0, mp_type_idx, W_mp, b_mp, agg1);

    // Stage 3a: QKV projection. 3072 tile-waves.
    k_qkv<<<3072 / 8, 256, 0, stream>>>(agg1, Wqkv, bqkv, qkvb);

    // Stage 3b: attention. 16384 threads.
    k_attn<<<(BB * HH * MM) / 256, 256, 0, stream>>>(qkvb, attno);

    // Stage 3c: output projection into d_out ([M,B,E]). 1024 tile-waves.
    k_oproj<<<1024 / 8, 256, 0, stream>>>(attno, Wo, bo, out);
}